// WSIGraphSAGE_4801773437763
// MI455X (gfx1250) — compile-verified
//
#include <hip/hip_runtime.h>
#include <hip/hip_bf16.h>

#define N_NODES   20000
#define N_EDGES   160000
#define IN_DIM    512
#define HID       256
#define N_CLASSES 2
#define NUM_GRAPHS 16

typedef __attribute__((ext_vector_type(2))) float v2f;
typedef __attribute__((ext_vector_type(8))) float v8f;

// ---------------------------------------------------------------- degree
__global__ void deg_kernel(const int* __restrict__ dst, float* __restrict__ deg) {
    int e = blockIdx.x * blockDim.x + threadIdx.x;
    if (e < N_EDGES) atomicAdd(&deg[dst[e]], 1.0f);
}

__global__ void invdeg_kernel(const float* __restrict__ deg, float* __restrict__ inv) {
    int i = blockIdx.x * blockDim.x + threadIdx.x;
    if (i < N_NODES) inv[i] = 1.0f / fmaxf(deg[i], 1.0f);
}

// ------------------------------------------------- edge scatter-add (sum)
template <int D>
__global__ void scatter_add_kernel(const float* __restrict__ feat,
                                   const int* __restrict__ src,
                                   const int* __restrict__ dst,
                                   float* __restrict__ aggr) {
    const int per_edge = D / 4;
    int tid = blockIdx.x * blockDim.x + threadIdx.x;
    if (tid >= N_EDGES * per_edge) return;
    int e = tid / per_edge;
    int c = (tid - e * per_edge) * 4;
    int s = src[e];
    int d = dst[e];
    const float4 v = *(const float4*)(feat + (size_t)s * D + c);
    float* p = aggr + (size_t)d * D + c;
    atomicAdd(p + 0, v.x);
    atomicAdd(p + 1, v.y);
    atomicAdd(p + 2, v.z);
    atomicAdd(p + 3, v.w);
}

// --------------------------------------------------------- fused SAGE GEMM
// out[M x 256] = relu( (aggr .* invdeg[:,None]) @ Wn + X @ Ws + bias )
// Block = 128 threads (4 waves). Block tile: 32 rows x 128 cols.
// Each wave owns a 32x32 register tile (2x2 sub-tiles of 16x16, 4 v8f accs),
// so each A-frag feeds 2 WMMAs and each B-frag feeds 2 WMMAs.
template <int K>
__launch_bounds__(128)
__global__ void sage_gemm_kernel(const float* __restrict__ Aagg,
                                 const float* __restrict__ Ax,
                                 const float* __restrict__ Wn,
                                 const float* __restrict__ Ws,
                                 const float* __restrict__ bias,
                                 const float* __restrict__ invd,
                                 float* __restrict__ out) {
    constexpr int LDSW = 68;   // stride 68 % 64 banks = 4 -> conflict-free b64 frag reads
    __shared__ float sAg[32][LDSW];
    __shared__ float sAx[32][LDSW];

    const int m0   = blockIdx.x * 32;
    const int wave = threadIdx.x >> 5;
    const int n0   = blockIdx.y * 128 + wave * 32;
    const int lane = threadIdx.x & 31;
    const int lm   = lane & 15;              // row-in-tile / column selector
    const int lh   = lane >> 4;              // half-wave selector

    v8f acc00 = {}, acc01 = {}, acc10 = {}, acc11 = {};

    for (int kc = 0; kc < K; kc += 64) {
        __syncthreads();
        // cooperative float4 fill: 32 rows x 64 K, both panels (4 granules/thread)
        for (int i = threadIdx.x; i < 32 * 16; i += 128) {
            int row  = i >> 4;
            int col  = (i & 15) * 4;
            int gidx = (m0 + row) * K + kc + col;
            const float4 vg = *(const float4*)(Aagg + gidx);
            const float4 vx = *(const float4*)(Ax + gidx);
            const float s = invd[m0 + row];               // fold mean-normalize
            *(float4*)&sAg[row][col] = make_float4(vg.x * s, vg.y * s, vg.z * s, vg.w * s);
            *(float4*)&sAx[row][col] = vx;
        }
        __syncthreads();

        #pragma unroll
        for (int kk = 0; kk < 64; kk += 4) {
            const int kb = kk + 2 * lh;      // A/B lane layout: K = 2*(lane/16)+j
            v2f ag0, ag1, ax0, ax1, bn0, bn1, bs0, bs1;
            ag0[0] = sAg[lm][kb];          ag0[1] = sAg[lm][kb + 1];
            ag1[0] = sAg[lm + 16][kb];     ag1[1] = sAg[lm + 16][kb + 1];
            ax0[0] = sAx[lm][kb];          ax0[1] = sAx[lm][kb + 1];
            ax1[0] = sAx[lm + 16][kb];     ax1[1] = sAx[lm + 16][kb + 1];
            const int gk = kc + kb;
            bn0[0] = Wn[gk * HID + n0 + lm];        bn0[1] = Wn[(gk + 1) * HID + n0 + lm];
            bn1[0] = Wn[gk * HID + n0 + 16 + lm];   bn1[1] = Wn[(gk + 1) * HID + n0 + 16 + lm];
            bs0[0] = Ws[gk * HID + n0 + lm];        bs0[1] = Ws[(gk + 1) * HID + n0 + lm];
            bs1[0] = Ws[gk * HID + n0 + 16 + lm];   bs1[1] = Ws[(gk + 1) * HID + n0 + 16 + lm];

            acc00 = __builtin_amdgcn_wmma_f32_16x16x4_f32(false, ag0, false, bn0, (short)0, acc00, false, false);
            acc00 = __builtin_amdgcn_wmma_f32_16x16x4_f32(false, ax0, false, bs0, (short)0, acc00, false, false);
            acc01 = __builtin_amdgcn_wmma_f32_16x16x4_f32(false, ag0, false, bn1, (short)0, acc01, false, false);
            acc01 = __builtin_amdgcn_wmma_f32_16x16x4_f32(false, ax0, false, bs1, (short)0, acc01, false, false);
            acc10 = __builtin_amdgcn_wmma_f32_16x16x4_f32(false, ag1, false, bn0, (short)0, acc10, false, false);
            acc10 = __builtin_amdgcn_wmma_f32_16x16x4_f32(false, ax1, false, bs0, (short)0, acc10, false, false);
            acc11 = __builtin_amdgcn_wmma_f32_16x16x4_f32(false, ag1, false, bn1, (short)0, acc11, false, false);
            acc11 = __builtin_amdgcn_wmma_f32_16x16x4_f32(false, ax1, false, bs1, (short)0, acc11, false, false);
        }
    }

    // epilogue: bias + ReLU.  C/D layout: VGPR v, lanes -> [M = v+8*lh][N = lm]
    const float bv0 = bias[n0 + lm];
    const float bv1 = bias[n0 + 16 + lm];
    #pragma unroll
    for (int v = 0; v < 8; ++v) {
        const int r0 = m0 + v + 8 * lh;
        const int r1 = r0 + 16;
        out[r0 * HID + n0 + lm]      = fmaxf(acc00[v] + bv0, 0.0f);
        out[r0 * HID + n0 + 16 + lm] = fmaxf(acc01[v] + bv1, 0.0f);
        out[r1 * HID + n0 + lm]      = fmaxf(acc10[v] + bv0, 0.0f);
        out[r1 * HID + n0 + 16 + lm] = fmaxf(acc11[v] + bv1, 0.0f);
    }
}

// ------------------------------------------------------------- mean pool
__global__ void cnt_kernel(const int* __restrict__ batch, float* __restrict__ cnt) {
    int i = blockIdx.x * blockDim.x + threadIdx.x;
    if (i < N_NODES) atomicAdd(&cnt[batch[i]], 1.0f);
}

__global__ void pool_kernel(const float* __restrict__ h,
                            const int* __restrict__ batch,
                            float* __restrict__ pooled) {
    int tid = blockIdx.x * blockDim.x + threadIdx.x;
    if (tid >= N_NODES * HID) return;
    int n = tid >> 8;          // / 256
    int c = tid & 255;
    atomicAdd(&pooled[batch[n] * HID + c], h[tid]);
}

// ------------------------------------------------------------------ MLP
__global__ void mlp_kernel(const float* __restrict__ pooled,
                           const float* __restrict__ cnt,
                           const float* __restrict__ fc1w,
                           const float* __restrict__ fc1b,
                           const float* __restrict__ fc2w,
                           const float* __restrict__ fc2b,
                           float* __restrict__ out) {
    __shared__ float pm[NUM_GRAPHS][HID];        // 16 KB
    __shared__ float hid[NUM_GRAPHS][HID / 2];   // 8 KB
    const int t = threadIdx.x;                   // 256 threads

    for (int i = t; i < NUM_GRAPHS * HID; i += 256) {
        int g = i >> 8;
        pm[g][i & 255] = pooled[i] / fmaxf(cnt[g], 1.0f);
    }
    __syncthreads();

    for (int i = t; i < NUM_GRAPHS * (HID / 2); i += 256) {
        int g = i >> 7;
        int j = i & 127;
        float s = fc1b[j];
        for (int k = 0; k < HID; ++k) s += pm[g][k] * fc1w[k * (HID / 2) + j];
        hid[g][j] = fmaxf(s, 0.0f);
    }
    __syncthreads();

    if (t < NUM_GRAPHS * N_CLASSES) {
        int g = t >> 1;
        int c = t & 1;
        float s = fc2b[c];
        for (int k = 0; k < HID / 2; ++k) s += hid[g][k] * fc2w[k * N_CLASSES + c];
        out[t] = s;
    }
}

// ---------------------------------------------------------------- launch
extern "C" void kernel_launch(void* const* d_in, const int* in_sizes, int n_in,
                              void* d_out, int out_size, void* d_ws, size_t ws_size,
                              hipStream_t stream) {
    (void)in_sizes; (void)n_in; (void)out_size; (void)ws_size;

    const float* x     = (const float*)d_in[0];
    const int*   ei    = (const int*)  d_in[1];
    const int*   batch = (const int*)  d_in[2];
    const float* W1n   = (const float*)d_in[3];
    const float* W1s   = (const float*)d_in[4];
    const float* b1    = (const float*)d_in[5];
    const float* W2n   = (const float*)d_in[6];
    const float* W2s   = (const float*)d_in[7];
    const float* b2    = (const float*)d_in[8];
    const float* fc1w  = (const float*)d_in[9];
    const float* fc1b  = (const float*)d_in[10];
    const float* fc2w  = (const float*)d_in[11];
    const float* fc2b  = (const float*)d_in[12];
    const int* src = ei;
    const int* dst = ei + N_EDGES;
    float* out = (float*)d_out;

    // workspace carve-up
    float* p     = (float*)d_ws;
    float* aggr1 = p; p += (size_t)N_NODES * IN_DIM;   // 41.0 MB
    float* h1    = p; p += (size_t)N_NODES * HID;      // 20.5 MB
    float* aggr2 = p; p += (size_t)N_NODES * HID;      // 20.5 MB
    float* h2    = p; p += (size_t)N_NODES * HID;      // 20.5 MB
    float* deg   = p; p += N_NODES;
    float* invd  = p; p += N_NODES;
    float* pooled= p; p += NUM_GRAPHS * HID;
    float* cnt   = p; p += NUM_GRAPHS;                 // contiguous after pooled

    hipMemsetAsync(aggr1, 0, (size_t)N_NODES * IN_DIM * sizeof(float), stream);
    hipMemsetAsync(aggr2, 0, (size_t)N_NODES * HID * sizeof(float), stream);
    hipMemsetAsync(deg,   0, N_NODES * sizeof(float), stream);
    hipMemsetAsync(pooled,0, (NUM_GRAPHS * HID + NUM_GRAPHS) * sizeof(float), stream);

    deg_kernel<<<(N_EDGES + 255) / 256, 256, 0, stream>>>(dst, deg);
    invdeg_kernel<<<(N_NODES + 255) / 256, 256, 0, stream>>>(deg, invd);

    // layer 1
    scatter_add_kernel<IN_DIM>
        <<<(N_EDGES * (IN_DIM / 4) + 255) / 256, 256, 0, stream>>>(x, src, dst, aggr1);
    sage_gemm_kernel<IN_DIM>
        <<<dim3(N_NODES / 32, HID / 128), 128, 0, stream>>>(aggr1, x, W1n, W1s, b1, invd, h1);

    // layer 2
    scatter_add_kernel<HID>
        <<<(N_EDGES * (HID / 4) + 255) / 256, 256, 0, stream>>>(h1, src, dst, aggr2);
    sage_gemm_kernel<HID>
        <<<dim3(N_NODES / 32, HID / 128), 128, 0, stream>>>(aggr2, h1, W2n, W2s, b2, invd, h2);

    // pool + MLP head
    cnt_kernel<<<(N_NODES + 255) / 256, 256, 0, stream>>>(batch, cnt);
    pool_kernel<<<(N_NODES * HID + 255) / 256, 256, 0, stream>>>(h2, batch, pooled);
    mlp_kernel<<<1, 256, 0, stream>>>(pooled, cnt, fc1w, fc1b, fc2w, fc2b, out);
}